// FeatureExtractor_46222438039690
// MI455X (gfx1250) — compile-verified
//
#include <hip/hip_runtime.h>
#include <math.h>

// ---------------------------------------------------------------------------
// FeatureExtractor for (T, 522, 3) landmarks -> (T, 39) features.
// Memory/latency-bound gather workload. gfx1250-specific data movement:
//   * TDM tensor_load_to_lds: one DMA per workgroup stages the 128-frame hand
//     tile (64 dwords/row, row stride 1566 dwords) into LDS, with TDM padding
//     (64dw interval, 4dw amount) producing a bank-friendly 68-dword stride.
//   * global_store_async_from_lds_b128: coalesced async output stores from an
//     LDS-staged (128 x 39)-float tile, tracked on ASYNCcnt.
// ---------------------------------------------------------------------------

namespace {
constexpr int   kN          = 522;
constexpr int   kFrameF     = kN * 3;     // 1566 floats per frame
constexpr int   kOutF       = 39;
constexpr float kEps        = 1e-8f;
constexpr int   kBlock      = 128;        // 4 waves (wave32)
constexpr int   kHandStride = 68;         // 64 data dwords + 4 TDM pad dwords
}  // namespace

typedef unsigned int v4u __attribute__((ext_vector_type(4)));
typedef int          v4i __attribute__((ext_vector_type(4)));
typedef int          v8i __attribute__((ext_vector_type(8)));

__device__ __forceinline__ float dot3(const float* a, const float* b) {
  return fmaf(a[0], b[0], fmaf(a[1], b[1], a[2] * b[2]));
}
__device__ __forceinline__ float len3(const float* a) { return sqrtf(dot3(a, a)); }

__device__ __forceinline__ float angle3(const float* a, const float* b) {
  float c = dot3(a, b) / (len3(a) * len3(b) + kEps);
  c = fminf(1.0f, fmaxf(-1.0f, c));
  return acosf(c);
}

// Palm center (mean of landmarks {0,1,5,9,13,17}) straight from global memory.
__device__ __forceinline__ void palm_center_global(const float* __restrict__ f,
                                                   float* __restrict__ c) {
  const int P[6] = {0, 1, 5, 9, 13, 17};
  float x = 0.f, y = 0.f, z = 0.f;
#pragma unroll
  for (int i = 0; i < 6; ++i) {
    const float* p = f + P[i] * 3;
    x += p[0]; y += p[1]; z += p[2];
  }
  c[0] = x * (1.0f / 6.0f);
  c[1] = y * (1.0f / 6.0f);
  c[2] = z * (1.0f / 6.0f);
}

__global__ __launch_bounds__(kBlock) void feat_kernel(
    const float* __restrict__ lm, float* __restrict__ out, int T) {
  __shared__ alignas(16) float s_hand[kBlock * kHandStride];
  __shared__ float s_center[kBlock][4];
  __shared__ alignas(16) float s_out[kBlock * kOutF];

  const int  tid        = threadIdx.x;
  const int  blockStart = blockIdx.x * kBlock;
  const int  t          = blockStart + tid;
  const bool active     = (t < T);
  const int  nact       = min(kBlock, T - blockStart);

  const float* frame = lm + (size_t)t * kFrameF;

  // ---- TDM: one tensor DMA per workgroup stages the hand tile into LDS.
  //      Tile: 64 dwords x nact rows, global row stride 1566 dwords, LDS rows
  //      padded by 4 dwords (interval code 5 = 64dw, amount code 3 = 4dw) so
  //      the LDS row stride is kHandStride = 68 dwords.
  if (tid < 32) {  // wave 0 issues the block-wide TDM op (EXEC is ignored)
    const unsigned           lds0 = (unsigned)(size_t)(void*)&s_hand[0];
    const unsigned long long ga =
        (unsigned long long)(size_t)(const void*)(lm + (size_t)blockStart * kFrameF);

    v4u g0;
    g0.x = 1u;                                             // count=1 (user D#)
    g0.y = lds0;                                           // lds_addr
    g0.z = (unsigned)ga;                                   // global_addr[31:0]
    g0.w = (unsigned)((ga >> 32) & 0x01FFFFFFull) | (2u << 30);  // [56:32]+type=2

    v8i g1;
    g1[0] = (int)((2u << 16)        // data_size = 4 bytes
                  | (1u << 20)      // pad_enable
                  | (5u << 22)      // pad_interval: 64 dwords
                  | (3u << 25));    // pad_amount:   4 dwords
    g1[1] = (int)(((unsigned)kFrameF & 0xFFFFu) << 16);    // tensor_dim0 lo
    g1[2] = (int)(((unsigned)nact & 0xFFFFu) << 16);       // tensor_dim1 lo
    g1[3] = (int)(64u << 16);                              // tile_dim0 = 64
    g1[4] = (int)(unsigned)nact;                           // tile_dim1; tile_dim2=0
    g1[5] = kFrameF;                                       // tensor_dim0_stride lo
    g1[6] = 0;                                             // stride hi bits
    g1[7] = 0;

    const v4i gz = {0, 0, 0, 0};
#if __clang_major__ >= 23
    const v8i gz8 = {0, 0, 0, 0, 0, 0, 0, 0};
    __builtin_amdgcn_tensor_load_to_lds(g0, g1, gz, gz, gz8, 0);
#else
    __builtin_amdgcn_tensor_load_to_lds(g0, g1, gz, gz, 0);
#endif
    __builtin_amdgcn_s_wait_tensorcnt(0);
  }
  __syncthreads();

  // ---- Pull the 21 hand points out of LDS and publish this frame's palm
  //      center so neighbor threads can build dc/d2c without HBM re-reads.
  float h[21][3];
  float c0[3] = {0.f, 0.f, 0.f};
  if (active) {
    const float* sh = &s_hand[tid * kHandStride];
#pragma unroll
    for (int p = 0; p < 21; ++p) {
      h[p][0] = sh[3 * p + 0];
      h[p][1] = sh[3 * p + 1];
      h[p][2] = sh[3 * p + 2];
    }
    const int P[6] = {0, 1, 5, 9, 13, 17};
    float x = 0.f, y = 0.f, z = 0.f;
#pragma unroll
    for (int i = 0; i < 6; ++i) {
      x += h[P[i]][0]; y += h[P[i]][1]; z += h[P[i]][2];
    }
    c0[0] = x * (1.0f / 6.0f);
    c0[1] = y * (1.0f / 6.0f);
    c0[2] = z * (1.0f / 6.0f);
    s_center[tid][0] = c0[0];
    s_center[tid][1] = c0[1];
    s_center[tid][2] = c0[2];
  }
  __syncthreads();

  if (active) {
    // ---- Finger flex: sum of two inter-phalanx angles per finger.
    float flex[5];
    {
      const int B[5] = {1, 5, 9, 13, 17};
#pragma unroll
      for (int f = 0; f < 5; ++f) {
        const int b = B[f];
        float v1[3], v2[3], v3[3];
#pragma unroll
        for (int k = 0; k < 3; ++k) {
          v1[k] = h[b + 1][k] - h[b + 0][k];
          v2[k] = h[b + 2][k] - h[b + 1][k];
          v3[k] = h[b + 3][k] - h[b + 2][k];
        }
        flex[f] = angle3(v1, v2) + angle3(v2, v3);
      }
    }

    // ---- Thumb opposition angle.
    float ta[3], tb[3];
#pragma unroll
    for (int k = 0; k < 3; ++k) {
      ta[k] = h[4][k] - h[2][k];
      tb[k] = h[5][k] - h[1][k];
    }
    const float thumb = angle3(ta, tb);

    // ---- Palm normal: normalized cross(h1-h0, h5-h0).
    float e1[3], e2[3], nrm[3];
#pragma unroll
    for (int k = 0; k < 3; ++k) {
      e1[k] = h[1][k] - h[0][k];
      e2[k] = h[5][k] - h[0][k];
    }
    nrm[0] = e1[1] * e2[2] - e1[2] * e2[1];
    nrm[1] = e1[2] * e2[0] - e1[0] * e2[2];
    nrm[2] = e1[0] * e2[1] - e1[1] * e2[0];
    const float ninv = 1.0f / (len3(nrm) + kEps);
    nrm[0] *= ninv; nrm[1] *= ninv; nrm[2] *= ninv;

    // ---- Center velocity / acceleration (neighbor centers from LDS; only
    //      the 2 boundary threads per block re-gather 6 points from global).
    float dc[3]  = {0.f, 0.f, 0.f};
    float d2c[3] = {0.f, 0.f, 0.f};
    float c1[3]  = {0.f, 0.f, 0.f};
    float c2[3]  = {0.f, 0.f, 0.f};
    if (t >= 1) {
      if (tid >= 1) {
        c1[0] = s_center[tid - 1][0];
        c1[1] = s_center[tid - 1][1];
        c1[2] = s_center[tid - 1][2];
      } else {
        palm_center_global(lm + (size_t)(t - 1) * kFrameF, c1);
      }
#pragma unroll
      for (int k = 0; k < 3; ++k) dc[k] = c0[k] - c1[k];
    }
    if (t >= 2) {
      if (tid >= 2) {
        c2[0] = s_center[tid - 2][0];
        c2[1] = s_center[tid - 2][1];
        c2[2] = s_center[tid - 2][2];
      } else {
        palm_center_global(lm + (size_t)(t - 2) * kFrameF, c2);
      }
#pragma unroll
      for (int k = 0; k < 3; ++k) d2c[k] = c0[k] - 2.0f * c1[k] + c2[k];
    }

    // ---- Gaze: ||pose[0] - pose[7]|| (landmarks 489 and 496).
    float g[3];
    {
      const float* pa = frame + 489 * 3;
      const float* pb = frame + 496 * 3;
#pragma unroll
      for (int k = 0; k < 3; ++k) g[k] = pa[k] - pb[k];
    }
    const float gaze = len3(g);

    // ---- Mouth aperture: ||max - min|| over 20 gathered face points.
    const int M[20] = {61, 185, 40, 39, 37, 0,  267, 269, 270, 409,
                       291, 375, 321, 405, 314, 17, 84,  181, 91,  146};
    float mn[3] = {3.4e38f, 3.4e38f, 3.4e38f};
    float mx[3] = {-3.4e38f, -3.4e38f, -3.4e38f};
#pragma unroll
    for (int i = 0; i < 20; ++i) {
      const float* q  = frame + (size_t)(21 + M[i]) * 3;
      const float qx = q[0], qy = q[1], qz = q[2];
      mn[0] = fminf(mn[0], qx); mx[0] = fmaxf(mx[0], qx);
      mn[1] = fminf(mn[1], qy); mx[1] = fmaxf(mx[1], qy);
      mn[2] = fminf(mn[2], qz); mx[2] = fmaxf(mx[2], qz);
    }
    float ap[3] = {mx[0] - mn[0], mx[1] - mn[1], mx[2] - mn[2]};
    const float aperture = len3(ap);

    // ---- Stage all 39 features into the LDS output tile.
    float* o = &s_out[tid * kOutF];
#pragma unroll
    for (int f = 0; f < 5; ++f) o[f] = flex[f];
    o[5] = thumb;
    o[6] = 0.f; o[7] = 0.f; o[8] = 0.f; o[9] = 0.f;
    o[10] = c0[0]; o[11] = c0[1]; o[12] = c0[2];
    o[13] = c0[0]; o[14] = c0[1]; o[15] = c0[2];
    o[16] = nrm[0]; o[17] = nrm[1]; o[18] = nrm[2];
    o[19] = nrm[0]; o[20] = nrm[1]; o[21] = nrm[2];
    o[22] = dc[0];  o[23] = dc[1];  o[24] = dc[2];
    o[25] = d2c[0]; o[26] = d2c[1]; o[27] = d2c[2];
    o[28] = 0.f; o[29] = 0.f; o[30] = 0.f;
    o[31] = 0.f; o[32] = 0.f; o[33] = 0.f;
    o[34] = gaze;
    o[35] = aperture;
    o[36] = 0.5f;
    o[37] = 0.f;
    o[38] = 0.f;
  }
  __syncthreads();

  // ---- Coalesced async stores: flush the block's contiguous (nact*39)-float
  //      output tile from LDS to global with b128 async stores (ASYNCcnt).
  {
    const int                totalDw = nact * kOutF;       // 4992 for full block
    const unsigned           ldsb = (unsigned)(size_t)(void*)&s_out[0];
    const unsigned long long gb =
        (unsigned long long)(size_t)(void*)(out + (size_t)blockStart * kOutF);
    for (int dw = tid * 4; dw + 4 <= totalDw; dw += kBlock * 4) {
      asm volatile("global_store_async_from_lds_b128 %0, %1, off"
                   :
                   : "v"(gb + (unsigned long long)dw * 4u), "v"(ldsb + (unsigned)dw * 4u)
                   : "memory");
    }
    const int tail = totalDw & ~3;                          // <4 dword remainder
    for (int dw = tail + tid; dw < totalDw; dw += kBlock) {
      asm volatile("global_store_async_from_lds_b32 %0, %1, off"
                   :
                   : "v"(gb + (unsigned long long)dw * 4u), "v"(ldsb + (unsigned)dw * 4u)
                   : "memory");
    }
  }
  asm volatile("s_wait_asynccnt 0" ::: "memory");
}

extern "C" void kernel_launch(void* const* d_in, const int* in_sizes, int n_in,
                              void* d_out, int out_size, void* d_ws, size_t ws_size,
                              hipStream_t stream) {
  (void)n_in; (void)out_size; (void)d_ws; (void)ws_size;
  const float* lm  = (const float*)d_in[0];
  float*       out = (float*)d_out;
  const int T    = in_sizes[0] / (kN * 3);
  const int grid = (T + kBlock - 1) / kBlock;
  feat_kernel<<<grid, kBlock, 0, stream>>>(lm, out, T);
}